// BarlowTwinsLoss_30288109372144
// MI455X (gfx1250) — compile-verified
//
#include <hip/hip_runtime.h>
#include <hip/hip_bf16.h>

// ---- types ----
typedef __attribute__((ext_vector_type(2))) float v2f;
typedef __attribute__((ext_vector_type(4))) float v4f;
typedef __attribute__((ext_vector_type(8))) float v8f;

#define D_COLS   512
#define N_STATS  5          // se, se2, st, st2, set
#define STAT_W   (D_COLS * N_STATS)   // 2560 columns of stats

// -----------------------------------------------------------------------------
// Phase 1: streaming per-column partial sums.
// Grid: NBLK blocks x 128 threads. Thread t owns columns 4t..4t+3 (one b128
// load per matrix per row). Grid-stride over rows; block writes its 2560
// partials (5 stats x 512 cols) to P[blk*2560 + stat*512 + col].
// Non-temporal loads: 256 MB streamed once, keep L2 clean.
// -----------------------------------------------------------------------------
__global__ __launch_bounds__(128)
void bt_partial_kernel(const float* __restrict__ e, const float* __restrict__ t,
                       float* __restrict__ P, int nblk, int nrows) {
    const int tid = threadIdx.x;           // 0..127
    const int col = tid << 2;              // 4 consecutive columns
    const int blk = blockIdx.x;

    v4f se  = {0.f, 0.f, 0.f, 0.f};
    v4f se2 = {0.f, 0.f, 0.f, 0.f};
    v4f st  = {0.f, 0.f, 0.f, 0.f};
    v4f st2 = {0.f, 0.f, 0.f, 0.f};
    v4f set = {0.f, 0.f, 0.f, 0.f};

    #pragma unroll 4
    for (int row = blk; row < nrows; row += nblk) {
        const size_t off = (size_t)row * D_COLS + col;
        const v4f ev = __builtin_nontemporal_load((const v4f*)(e + off));
        const v4f tv = __builtin_nontemporal_load((const v4f*)(t + off));
        se  += ev;
        st  += tv;
        se2 += ev * ev;
        st2 += tv * tv;
        set += ev * tv;
    }

    float* base = P + (size_t)blk * STAT_W;
    #pragma unroll
    for (int j = 0; j < 4; ++j) {
        base[0 * D_COLS + col + j] = se[j];
        base[1 * D_COLS + col + j] = se2[j];
        base[2 * D_COLS + col + j] = st[j];
        base[3 * D_COLS + col + j] = st2[j];
        base[4 * D_COLS + col + j] = set[j];
    }
}

// -----------------------------------------------------------------------------
// Phase 2: cross-block reduction via V_WMMA_F32_16X16X4_F32.
// A = 16x4 all-ones, B = 4x16 chunk of partial rows, C accumulates.
// D[m,n] = sum_k B[k,n] + C[m,n]  -> every row of C holds the column sums,
// so component 0 of lanes 0..15 is the reduced value for column lane&15.
// With a ones-A, any bijective assignment of the 4 K-rows onto
// (component, lane-half) is correct — K-order is irrelevant to the sum.
// One wave (32 threads, EXEC all ones) per 16-column group; 160 groups.
// -----------------------------------------------------------------------------
__global__ __launch_bounds__(32)
void bt_reduce_wmma_kernel(const float* __restrict__ P, float* __restrict__ stats,
                           int nblk) {
    const int lane = threadIdx.x;             // 0..31, full wave32
    const int g    = blockIdx.x;              // 0..159
    const int col  = g * 16 + (lane & 15);
    const int h    = lane >> 4;               // lane half selects 2 of 4 K-rows

    v2f a; a[0] = 1.0f; a[1] = 1.0f;          // 16x4 ones matrix
    v8f c = {0.f, 0.f, 0.f, 0.f, 0.f, 0.f, 0.f, 0.f};

    const float* p0 = P + col;
    for (int r = 0; r < nblk; r += 4) {
        v2f b;
        const size_t r0 = (size_t)(r + 2 * h);
        b[0] = p0[r0 * STAT_W];
        b[1] = p0[(r0 + 1) * STAT_W];
        c = __builtin_amdgcn_wmma_f32_16x16x4_f32(
                /*neg_a=*/false, a, /*neg_b=*/false, b,
                /*c_mod=*/(short)0, c, /*reuse_a=*/false, /*reuse_b=*/false);
    }

    if (lane < 16) stats[col] = c[0];
}

// -----------------------------------------------------------------------------
// Phase 3: closed-form per-column loss term + 512 -> 1 LDS tree reduction.
// -----------------------------------------------------------------------------
__global__ __launch_bounds__(512)
void bt_finalize_kernel(const float* __restrict__ stats, float* __restrict__ out,
                        int nrows) {
    __shared__ float red[D_COLS];
    const int d = threadIdx.x;                // 0..511

    const float B   = (float)nrows;
    const float se  = stats[0 * D_COLS + d];
    const float se2 = stats[1 * D_COLS + d];
    const float st  = stats[2 * D_COLS + d];
    const float st2 = stats[3 * D_COLS + d];
    const float sep = stats[4 * D_COLS + d];

    const float vare = (se2 - se * se / B) / (B - 1.0f);
    const float vart = (st2 - st * st / B) / (B - 1.0f);
    const float stde = sqrtf(fmaxf(vare, 0.0f)) + 1e-9f;
    const float stdt = sqrtf(fmaxf(vart, 0.0f)) + 1e-9f;
    const float cov  = sep - se * st / B;
    const float cd   = cov / (stde * stdt) / B;
    const float v    = (1.0f - cd) * (1.0f - cd);

    red[d] = v;
    __syncthreads();
    #pragma unroll
    for (int s = D_COLS / 2; s > 0; s >>= 1) {
        if (d < s) red[d] += red[d + s];
        __syncthreads();
    }
    if (d == 0) out[0] = red[0];
}

// -----------------------------------------------------------------------------
extern "C" void kernel_launch(void* const* d_in, const int* in_sizes, int n_in,
                              void* d_out, int out_size, void* d_ws, size_t ws_size,
                              hipStream_t stream) {
    const float* e   = (const float*)d_in[0];
    const float* tau = (const float*)d_in[1];
    float* out       = (float*)d_out;

    const int nrows = in_sizes[0] / D_COLS;   // 65536 for the reference shapes

    // Workspace layout: [stats: 2560 f32][partials: NBLK * 2560 f32]
    float* stats    = (float*)d_ws;
    float* partials = stats + STAT_W;

    long cap_rows = ((long)(ws_size / sizeof(float)) - STAT_W) / STAT_W;
    int nblk = 512;                            // 2048 waves of streaming work
    if (cap_rows < nblk) nblk = (int)cap_rows;
    nblk &= ~3;                                // multiple of 4 for WMMA K-chunks
    if (nblk < 4) nblk = 4;

    bt_partial_kernel<<<nblk, 128, 0, stream>>>(e, tau, partials, nblk, nrows);
    bt_reduce_wmma_kernel<<<STAT_W / 16, 32, 0, stream>>>(partials, stats, nblk);
    bt_finalize_kernel<<<1, D_COLS, 0, stream>>>(stats, out, nrows);
}